// DRGCN_64149631533439
// MI455X (gfx1250) — compile-verified
//
#include <hip/hip_runtime.h>
#include <hip/hip_bf16.h>

typedef __attribute__((ext_vector_type(2))) float v2f;
typedef __attribute__((ext_vector_type(8))) float v8f;
typedef __attribute__((ext_vector_type(4))) unsigned u32x4;
typedef __attribute__((ext_vector_type(8))) int i32x8;
typedef __attribute__((ext_vector_type(4))) int i32x4;

__device__ __forceinline__ v8f wmma4(v2f a, v2f b, v8f c) {
  // V_WMMA_F32_16X16X4_F32 : exact fp32 D = A(16x4) * B(4x16) + C(16x16)
  return __builtin_amdgcn_wmma_f32_16x16x4_f32(false, a, false, b, (short)0, c,
                                               false, false);
}

// ------------------------- TDM: 1-D contiguous DMA global -> LDS (per §8 D#)
__device__ __forceinline__ void tdm_load_1d(const float* gsrc, float* ldst,
                                            int nElem) {
  unsigned lds = (unsigned)(size_t)ldst;  // LDS aperture: low 32 bits = offset
  unsigned long long ga = (unsigned long long)(size_t)gsrc;
  unsigned dim0 = (unsigned)nElem;
  u32x4 g0;
  g0[0] = 1u;                                        // count=1, user mode
  g0[1] = lds;                                       // lds_addr (bytes)
  g0[2] = (unsigned)(ga & 0xFFFFFFFFu);              // global_addr[31:0]
  g0[3] = (unsigned)((ga >> 32) & 0x01FFFFFFu)       // global_addr[56:32]
          | (2u << 30);                              // type = 2 ("image")
  i32x8 g1;
  g1[0] = (int)(2u << 16);                           // data_size = 4B
  g1[1] = (int)((dim0 & 0xFFFFu) << 16);             // tensor_dim0[15:0]
  g1[2] = (int)(((dim0 >> 16) & 0xFFFFu) | (1u << 16)); // dim0 hi | tensor_dim1=1
  g1[3] = (int)((dim0 & 0xFFFFu) << 16);             // tile_dim0 = nElem
  g1[4] = 1;                                         // tile_dim1 = 1
  g1[5] = (int)dim0;                                 // tensor_dim0_stride lo
  g1[6] = 0;
  g1[7] = 0;
  i32x4 z = {0, 0, 0, 0};
#if __clang_major__ >= 23
  i32x8 z8 = {0, 0, 0, 0, 0, 0, 0, 0};
  __builtin_amdgcn_tensor_load_to_lds(g0, g1, z, z, z8, 0);
#else
  __builtin_amdgcn_tensor_load_to_lds(g0, g1, z, z, 0);
#endif
}

// ---------------------------------------------------------------- utilities
__global__ void zero_f32(float* __restrict__ p, long long n) {
  long long i = (long long)blockIdx.x * blockDim.x + threadIdx.x;
  long long st = (long long)gridDim.x * blockDim.x;
  for (; i < n; i += st) p[i] = 0.f;
}

// column sums of h (N x d), d in {256,512}, blockDim=256
__global__ __launch_bounds__(256)
void col_sum(const float* __restrict__ h, float* __restrict__ msum, int n, int d) {
  int rowsPerBlock = (n + gridDim.x - 1) / gridDim.x;
  int r0 = blockIdx.x * rowsPerBlock;
  int r1 = min(n, r0 + rowsPerBlock);
  int c0 = threadIdx.x, c1 = threadIdx.x + 256;
  float a0 = 0.f, a1 = 0.f;
  for (int r = r0; r < r1; ++r) {
    const float* row = &h[(size_t)r * d];
    a0 += row[c0];
    if (c1 < d) a1 += row[c1];
  }
  if (r0 < r1) {
    atomicAdd(&msum[c0], a0);
    if (c1 < d) atomicAdd(&msum[c1], a1);
  }
}

// ------------------------------------------------- Gram matrix G = h^T h (d x d)
__global__ __launch_bounds__(256)
void ata_wmma(const float* __restrict__ A, float* __restrict__ G, int N, int d) {
  int waveId = blockIdx.x * (blockDim.x >> 5) + (threadIdx.x >> 5);
  int lane = threadIdx.x & 31;
  int tiles = d >> 4;
  if (waveId >= tiles * tiles) return;          // uniform per wave
  int tm = waveId / tiles, tn = waveId % tiles;
  int i0 = tm << 4, j0 = tn << 4;
  int m = lane & 15;
  int kh = (lane >> 4) << 1;                    // 0 or 2
  v8f acc = {};
  for (int k = 0; k < N; k += 4) {
    const float* r0 = &A[(size_t)(k + kh) * d];
    const float* r1 = r0 + d;
    v2f a = { r0[i0 + m], r1[i0 + m] };         // A^T fragment
    v2f b = { r0[j0 + m], r1[j0 + m] };
    acc = wmma4(a, b, acc);
  }
  int rb = i0 + ((lane >> 4) << 3);
#pragma unroll
  for (int i2 = 0; i2 < 8; ++i2)
    G[(size_t)(rb + i2) * d + j0 + m] = acc[i2];
}

// ------------------------------- top-6 eigenvectors of C = G - N m m^T (one block)
__device__ __forceinline__ float bsum512(float v, float* red) {
  int t = threadIdx.x;
  red[t] = v; __syncthreads();
#pragma unroll
  for (int s = 256; s > 0; s >>= 1) {
    if (t < s) red[t] += red[t + s];
    __syncthreads();
  }
  float r = red[0]; __syncthreads();
  return r;
}

__global__ __launch_bounds__(512)
void topk_eig(const float* __restrict__ G, const float* __restrict__ msum,
              int d, float Nf, float* __restrict__ Vs, float* __restrict__ offv) {
  __shared__ float Q[512 * 6];
  __shared__ float Y[512 * 6];
  __shared__ float red[512];
  __shared__ float sZ[6];
  __shared__ float sB[6][6];
  __shared__ float sR[6][6];
  __shared__ float sW[6];
  int t = threadIdx.x;
  bool act = t < d;
  float m = act ? msum[t] / Nf : 0.f;
  if (act) {
#pragma unroll
    for (int c = 0; c < 6; ++c) {
      unsigned u = (unsigned)(t * 6 + c) * 2654435761u + 0x9e3779b9u;
      u ^= u >> 16; u *= 0x85ebca6bu; u ^= u >> 13;
      Q[t * 6 + c] = (float)(u & 0xFFFFu) * (1.f / 65536.f) - 0.5f;
    }
  }
  __syncthreads();
  for (int it = 0; it <= 40; ++it) {
    for (int c = 0; c < 6; ++c) {
      float z = bsum512(act ? m * Q[t * 6 + c] : 0.f, red);
      if (t == 0) sZ[c] = z;
    }
    __syncthreads();
    if (act) {
      float a0 = 0, a1 = 0, a2 = 0, a3 = 0, a4 = 0, a5 = 0;
      const float* g = &G[(size_t)t * d];
      for (int j = 0; j < d; ++j) {
        float gv = g[j];
        a0 += gv * Q[j * 6 + 0]; a1 += gv * Q[j * 6 + 1];
        a2 += gv * Q[j * 6 + 2]; a3 += gv * Q[j * 6 + 3];
        a4 += gv * Q[j * 6 + 4]; a5 += gv * Q[j * 6 + 5];
      }
      float nm = Nf * m;
      Y[t * 6 + 0] = a0 - nm * sZ[0]; Y[t * 6 + 1] = a1 - nm * sZ[1];
      Y[t * 6 + 2] = a2 - nm * sZ[2]; Y[t * 6 + 3] = a3 - nm * sZ[3];
      Y[t * 6 + 4] = a4 - nm * sZ[4]; Y[t * 6 + 5] = a5 - nm * sZ[5];
    }
    __syncthreads();
    if (it == 40) break;
    // modified Gram-Schmidt: Q <- orth(Y)
    for (int c = 0; c < 6; ++c) {
      for (int p = 0; p < c; ++p) {
        float dp = bsum512(act ? Y[t * 6 + c] * Q[t * 6 + p] : 0.f, red);
        if (act) Y[t * 6 + c] -= dp * Q[t * 6 + p];
        __syncthreads();
      }
      float nr = bsum512(act ? Y[t * 6 + c] * Y[t * 6 + c] : 0.f, red);
      float inv = rsqrtf(fmaxf(nr, 1e-30f));
      if (act) Q[t * 6 + c] = Y[t * 6 + c] * inv;
      __syncthreads();
    }
  }
  // Rayleigh-Ritz: B = Q^T (C Q)
  for (int p = 0; p < 6; ++p)
    for (int c = 0; c < 6; ++c) {
      float b = bsum512(act ? Q[t * 6 + p] * Y[t * 6 + c] : 0.f, red);
      if (t == 0) sB[p][c] = b;
    }
  __syncthreads();
  if (t == 0) {
    float A[6][6], R[6][6];
    for (int p = 0; p < 6; ++p)
      for (int c = 0; c < 6; ++c) {
        A[p][c] = 0.5f * (sB[p][c] + sB[c][p]);
        R[p][c] = (p == c) ? 1.f : 0.f;
      }
    for (int sweep = 0; sweep < 16; ++sweep)
      for (int p = 0; p < 5; ++p)
        for (int q = p + 1; q < 6; ++q) {
          float apq = A[p][q];
          if (fabsf(apq) < 1e-12f) continue;
          float th = 0.5f * (A[q][q] - A[p][p]) / apq;
          float tt = (th >= 0.f ? 1.f : -1.f) / (fabsf(th) + sqrtf(th * th + 1.f));
          float cc = rsqrtf(tt * tt + 1.f);
          float ss = tt * cc;
          for (int k = 0; k < 6; ++k) {
            float akp = A[k][p], akq = A[k][q];
            A[k][p] = cc * akp - ss * akq;
            A[k][q] = ss * akp + cc * akq;
          }
          for (int k = 0; k < 6; ++k) {
            float apk = A[p][k], aqk = A[q][k];
            A[p][k] = cc * apk - ss * aqk;
            A[q][k] = ss * apk + cc * aqk;
          }
          for (int k = 0; k < 6; ++k) {
            float rkp = R[k][p], rkq = R[k][q];
            R[k][p] = cc * rkp - ss * rkq;
            R[k][q] = ss * rkp + cc * rkq;
          }
        }
    int idx[6];
    for (int c = 0; c < 6; ++c) idx[c] = c;
    for (int c = 0; c < 6; ++c) {            // sort eigenvalues descending
      int best = c;
      for (int q = c + 1; q < 6; ++q)
        if (A[idx[q]][idx[q]] > A[idx[best]][idx[best]]) best = q;
      int tmp = idx[c]; idx[c] = idx[best]; idx[best] = tmp;
    }
    for (int c = 0; c < 6; ++c) {
      sW[c] = A[idx[c]][idx[c]];
      for (int k = 0; k < 6; ++k) sR[k][c] = R[k][idx[c]];
    }
  }
  __syncthreads();
  float vloc[6];
  if (act) {
#pragma unroll
    for (int c = 0; c < 6; ++c) {
      float v = 0.f;
#pragma unroll
      for (int p = 0; p < 6; ++p) v += Q[t * 6 + p] * sR[p][c];
      vloc[c] = v;
    }
  } else {
#pragma unroll
    for (int c = 0; c < 6; ++c) vloc[c] = 0.f;
  }
  for (int c = 0; c < 6; ++c) {
    float invs = rsqrtf(fmaxf(sW[c], 1e-12f));
    float dotm = bsum512(act ? m * vloc[c] : 0.f, red);
    if (t == 0) offv[c] = dotm * invs;
    if (act) Vs[t * 6 + c] = vloc[c] * invs;
  }
}

// -------------------------------- loc = h @ Vs - off ; sq = |loc|^2 (1 wave / node)
__global__ __launch_bounds__(256)
void project_loc(const float* __restrict__ h, const float* __restrict__ Vs,
                 const float* __restrict__ offv, float* __restrict__ loc,
                 float* __restrict__ sqv, int n, int d) {
  int wid = (int)(((size_t)blockIdx.x * blockDim.x + threadIdx.x) >> 5);
  int lane = threadIdx.x & 31;
  if (wid >= n) return;
  const float* hp = &h[(size_t)wid * d];
  float a0 = 0, a1 = 0, a2 = 0, a3 = 0, a4 = 0, a5 = 0;
  for (int x = lane; x < d; x += 32) {
    float hv = hp[x];
    const float* vp = &Vs[x * 6];
    a0 += hv * vp[0]; a1 += hv * vp[1]; a2 += hv * vp[2];
    a3 += hv * vp[3]; a4 += hv * vp[4]; a5 += hv * vp[5];
  }
#pragma unroll
  for (int s = 16; s > 0; s >>= 1) {
    a0 += __shfl_xor(a0, s, 32); a1 += __shfl_xor(a1, s, 32);
    a2 += __shfl_xor(a2, s, 32); a3 += __shfl_xor(a3, s, 32);
    a4 += __shfl_xor(a4, s, 32); a5 += __shfl_xor(a5, s, 32);
  }
  if (lane == 0) {
    float v0 = a0 - offv[0], v1 = a1 - offv[1], v2 = a2 - offv[2];
    float v3 = a3 - offv[3], v4 = a4 - offv[4], v5 = a5 - offv[5];
    float* lp = &loc[(size_t)wid * 6];
    lp[0] = v0; lp[1] = v1; lp[2] = v2; lp[3] = v3; lp[4] = v4; lp[5] = v5;
    sqv[wid] = v0 * v0 + v1 * v1 + v2 * v2 + v3 * v3 + v4 * v4 + v5 * v5;
  }
}

// ------------------- brute-force 5-NN, TDM double-buffered LDS tile streaming
#define KTILE 1024
__global__ __launch_bounds__(256)
void knn_kernel(const float* __restrict__ loc, const float* __restrict__ sqv,
                int n, int* __restrict__ nbr) {
  __shared__ __align__(16) float sp[2][KTILE * 6];
  __shared__ __align__(16) float ssq[2][KTILE];
  int i = blockIdx.x * 256 + threadIdx.x;
  bool act = i < n;
  float q0 = 0, q1 = 0, q2 = 0, q3 = 0, q4 = 0, q5 = 0, sqi = 0;
  if (act) {
    const float* qp = &loc[(size_t)i * 6];
    q0 = qp[0]; q1 = qp[1]; q2 = qp[2]; q3 = qp[3]; q4 = qp[4]; q5 = qp[5];
    sqi = sqv[i];
  }
  float bd0 = 3.4e38f, bd1 = 3.4e38f, bd2 = 3.4e38f, bd3 = 3.4e38f, bd4 = 3.4e38f;
  int bi0 = 0, bi1 = 0, bi2 = 0, bi3 = 0, bi4 = 0;
  int nTiles = (n + KTILE - 1) / KTILE;
  // prologue: wave 0 kicks the TDM for tile 0
  if (threadIdx.x < 32) {
    int c0 = min(KTILE, n);
    tdm_load_1d(loc, &sp[0][0], c0 * 6);
    tdm_load_1d(sqv, &ssq[0][0], c0);
  }
  for (int ti = 0; ti < nTiles; ++ti) {
    int t0 = ti * KTILE;
    int cnt = min(KTILE, n - t0);
    int buf = ti & 1;
    __syncthreads();            // readers of buf^1 (tile ti-1 consumers) are done
    if (threadIdx.x < 32) {
      if (ti + 1 < nTiles) {
        int t1 = t0 + KTILE;
        int cnt1 = min(KTILE, n - t1);
        tdm_load_1d(&loc[(size_t)t1 * 6], &sp[buf ^ 1][0], cnt1 * 6);
        tdm_load_1d(&sqv[t1], &ssq[buf ^ 1][0], cnt1);
        // in-order TENSORcnt: <=2 means tile ti landed, tile ti+1 still in flight
        __builtin_amdgcn_s_wait_tensorcnt(2);
      } else {
        __builtin_amdgcn_s_wait_tensorcnt(0);
      }
    }
    __syncthreads();            // publish: buf is ready for all waves
    if (act) {
      const float* spb = &sp[buf][0];
      const float* sqb = &ssq[buf][0];
      for (int j = 0; j < cnt; ++j) {
        const float* pj = &spb[j * 6];
        float dot = q0 * pj[0] + q1 * pj[1] + q2 * pj[2] +
                    q3 * pj[3] + q4 * pj[4] + q5 * pj[5];
        float dist = sqi - 2.f * dot + sqb[j];
        int jj = t0 + j;
        if (jj != i && dist < bd4) {
          if (dist < bd0) {
            bd4 = bd3; bi4 = bi3; bd3 = bd2; bi3 = bi2; bd2 = bd1; bi2 = bi1;
            bd1 = bd0; bi1 = bi0; bd0 = dist; bi0 = jj;
          } else if (dist < bd1) {
            bd4 = bd3; bi4 = bi3; bd3 = bd2; bi3 = bi2; bd2 = bd1; bi2 = bi1;
            bd1 = dist; bi1 = jj;
          } else if (dist < bd2) {
            bd4 = bd3; bi4 = bi3; bd3 = bd2; bi3 = bi2; bd2 = dist; bi2 = jj;
          } else if (dist < bd3) {
            bd4 = bd3; bi4 = bi3; bd3 = dist; bi3 = jj;
          } else {
            bd4 = dist; bi4 = jj;
          }
        }
      }
    }
  }
  if (act) {
    int* o = &nbr[(size_t)i * 5];
    o[0] = bi0; o[1] = bi1; o[2] = bi2; o[3] = bi3; o[4] = bi4;
  }
}

// --------------------------------------- C(MxNc) = A(MxK) @ B(KxNc), fp32 WMMA
__global__ __launch_bounds__(256)
void gemm_wmma(const float* __restrict__ A, const float* __restrict__ B,
               float* __restrict__ C, int M, int K, int Nc) {
  int waveId = blockIdx.x * (blockDim.x >> 5) + (threadIdx.x >> 5);
  int lane = threadIdx.x & 31;
  int tilesN = Nc >> 6;                  // 64-wide wave tile (4 fragments)
  if (waveId >= (M >> 4) * tilesN) return;   // uniform per wave
  int tm = waveId / tilesN, tn = waveId % tilesN;
  int m0 = tm << 4, n0 = tn << 6;
  int row = m0 + (lane & 15);
  int kh = (lane >> 4) << 1;             // 0 or 2
  int col = n0 + (lane & 15);
  v8f acc0 = {}, acc1 = {}, acc2 = {}, acc3 = {};
  for (int k = 0; k < K; k += 4) {
    v2f a = *(const v2f*)&A[(size_t)row * K + k + kh];
    const float* bp0 = &B[(size_t)(k + kh) * Nc + col];
    const float* bp1 = bp0 + Nc;
    v2f b0 = { bp0[0],  bp1[0]  };
    v2f b1 = { bp0[16], bp1[16] };
    v2f b2 = { bp0[32], bp1[32] };
    v2f b3 = { bp0[48], bp1[48] };
    acc0 = wmma4(a, b0, acc0);
    acc1 = wmma4(a, b1, acc1);
    acc2 = wmma4(a, b2, acc2);
    acc3 = wmma4(a, b3, acc3);
  }
  int rb = m0 + ((lane >> 4) << 3);
#pragma unroll
  for (int i2 = 0; i2 < 8; ++i2) {
    size_t o = (size_t)(rb + i2) * Nc + col;
    C[o]      = acc0[i2];
    C[o + 16] = acc1[i2];
    C[o + 32] = acc2[i2];
    C[o + 48] = acc3[i2];
  }
}

// ------------------------------ base edges: agg[dst] += hr0[src] (atomic, f32x4)
__global__ __launch_bounds__(256)
void scatter_base(const int* __restrict__ src, const int* __restrict__ dst,
                  const float* __restrict__ hr, float* __restrict__ agg,
                  long long total, int H) {
  long long tid = (long long)blockIdx.x * blockDim.x + threadIdx.x;
  if (tid >= total) return;
  int chunks = H >> 2;
  int e = (int)(tid / chunks);
  int c4 = (int)(tid % chunks) << 2;
  int s = src[e], d2 = dst[e];
  float4 v = *(const float4*)&hr[(size_t)s * H + c4];
  float* o = &agg[(size_t)d2 * H + c4];
  atomicAdd(o + 0, v.x); atomicAdd(o + 1, v.y);
  atomicAdd(o + 2, v.z); atomicAdd(o + 3, v.w);
}

// -------------------- KNN edges: dst = i is dense -> atomic-free gather-sum
__global__ __launch_bounds__(256)
void gather_knn(const int* __restrict__ nbr, const float* __restrict__ hr1,
                float* __restrict__ agg, int H) {
  int i = blockIdx.x, c = threadIdx.x;
  const int* nb = &nbr[(size_t)i * 5];
  float s = 0.f;
#pragma unroll
  for (int k = 0; k < 5; ++k) s += hr1[(size_t)nb[k] * H + c];
  agg[(size_t)i * H + c] += s;
}

// ---------------- LayerNorm(agg)*s+b + bias + h@Wloop (+ReLU), 1 block / node
__global__ __launch_bounds__(256)
void ln_finish(const float* __restrict__ agg, const float* __restrict__ hloop,
               const float* __restrict__ bias, const float* __restrict__ lns,
               const float* __restrict__ lnb, float* __restrict__ out,
               int H, int relu) {
  __shared__ float red[256];
  __shared__ float sMu, sVar;
  int i = blockIdx.x, t = threadIdx.x;
  float a = agg[(size_t)i * H + t];
  red[t] = a; __syncthreads();
  for (int s = 128; s > 0; s >>= 1) {
    if (t < s) red[t] += red[t + s];
    __syncthreads();
  }
  if (t == 0) sMu = red[0] / (float)H;
  __syncthreads();
  float dv = a - sMu;
  red[t] = dv * dv; __syncthreads();
  for (int s = 128; s > 0; s >>= 1) {
    if (t < s) red[t] += red[t + s];
    __syncthreads();
  }
  if (t == 0) sVar = red[0] / (float)H;
  __syncthreads();
  float o = dv * rsqrtf(sVar + 1e-5f) * lns[t] + lnb[t] + bias[t] +
            hloop[(size_t)i * H + t];
  if (relu) o = fmaxf(o, 0.f);
  out[(size_t)i * H + t] = o;
}

// =========================================================== host entry point
extern "C" void kernel_launch(void* const* d_in, const int* in_sizes, int n_in,
                              void* d_out, int out_size, void* d_ws, size_t ws_size,
                              hipStream_t stream) {
  (void)n_in; (void)out_size; (void)ws_size;
  const float* inputs = (const float*)d_in[0];
  const int*   src0   = (const int*)d_in[1];
  const int*   dst0   = (const int*)d_in[2];
  const float* W0     = (const float*)d_in[3];   // (2, 512, 256)
  const float* loop0  = (const float*)d_in[4];
  const float* b0     = (const float*)d_in[5];
  const float* lns0   = (const float*)d_in[6];
  const float* lnb0   = (const float*)d_in[7];
  const float* W1     = (const float*)d_in[8];   // (2, 256, 256)
  const float* loop1  = (const float*)d_in[9];
  const float* b1     = (const float*)d_in[10];
  const float* lns1   = (const float*)d_in[11];
  const float* lnb1   = (const float*)d_in[12];

  const int Nn = 50000, Din = 512, H = 256;
  const int E0 = in_sizes[1];

  float* wsf = (float*)d_ws;
  size_t off = 0;
  float* mean = wsf + off; off += 512;
  float* G    = wsf + off; off += 512 * 512;
  float* Vs   = wsf + off; off += 512 * 6;
  float* offv = wsf + off; off += 8;
  float* loc  = wsf + off; off += (size_t)Nn * 6;
  float* sqv  = wsf + off; off += Nn;
  int*   nbr  = (int*)(wsf + off); off += (size_t)Nn * 5;
  float* hr0  = wsf + off; off += (size_t)Nn * H;
  float* hr1  = wsf + off; off += (size_t)Nn * H;
  float* hlp  = wsf + off; off += (size_t)Nn * H;
  float* agg  = wsf + off; off += (size_t)Nn * H;
  float* h1   = wsf + off; off += (size_t)Nn * H;

  auto layer = [&](const float* h, int d, const float* Wr, const float* Wl,
                   const float* bias, const float* lns, const float* lnb,
                   int relu, float* out) {
    // --- PCA(top-6) of h ---
    zero_f32<<<1, 512, 0, stream>>>(mean, d);
    col_sum<<<400, 256, 0, stream>>>(h, mean, Nn, d);
    {
      int tiles = d >> 4;
      int waves = tiles * tiles;
      int blocks = (waves * 32 + 255) / 256;
      ata_wmma<<<blocks, 256, 0, stream>>>(h, G, Nn, d);
    }
    topk_eig<<<1, 512, 0, stream>>>(G, mean, d, (float)Nn, Vs, offv);
    project_loc<<<((size_t)Nn * 32 + 255) / 256, 256, 0, stream>>>(
        h, Vs, offv, loc, sqv, Nn, d);
    // --- dynamic 5-NN graph (TDM-streamed tiles) ---
    knn_kernel<<<(Nn + 255) / 256, 256, 0, stream>>>(loc, sqv, Nn, nbr);
    // --- per-relation GEMMs (fp32 WMMA) ---
    {
      int waves = (Nn >> 4) * (H >> 6);
      int blocks = (waves * 32 + 255) / 256;
      gemm_wmma<<<blocks, 256, 0, stream>>>(h, Wr,                 hr0, Nn, d, H);
      gemm_wmma<<<blocks, 256, 0, stream>>>(h, Wr + (size_t)d * H, hr1, Nn, d, H);
      gemm_wmma<<<blocks, 256, 0, stream>>>(h, Wl,                 hlp, Nn, d, H);
    }
    // --- message aggregation ---
    zero_f32<<<2048, 256, 0, stream>>>(agg, (long long)Nn * H);
    {
      long long total = (long long)E0 * (H >> 2);
      int blocks = (int)((total + 255) / 256);
      scatter_base<<<blocks, 256, 0, stream>>>(src0, dst0, hr0, agg, total, H);
    }
    gather_knn<<<Nn, 256, 0, stream>>>(nbr, hr1, agg, H);
    // --- LayerNorm + bias + self-loop (+ReLU) ---
    ln_finish<<<Nn, 256, 0, stream>>>(agg, hlp, bias, lns, lnb, out, H, relu);
  };

  layer(inputs, Din, W0, loop0, b0, lns0, lnb0, 1, h1);
  layer(h1,     H,   W1, loop1, b1, lns1, lnb1, 0, (float*)d_out);
}